// GFMLayer_2851858285039
// MI455X (gfx1250) — compile-verified
//
#include <hip/hip_runtime.h>
#include <hip/hip_bf16.h>

#define NU 200000
#define NI 100000
#define DD 128
#define NE 600000
#define EPS 1e-5f
#define LDP 132   // padded LDS row stride (dwords): (4*M + c) % 64 distinct -> conflict-free

typedef __attribute__((ext_vector_type(2))) float v2f;
typedef __attribute__((ext_vector_type(8))) float v8f;

__device__ __forceinline__ void atomAddF(float* p, float v) {
    __hip_atomic_fetch_add(p, v, __ATOMIC_RELAXED, __HIP_MEMORY_SCOPE_AGENT);
}

// ---------------------------------------------------------------- zero scratch
__global__ void k_zero(float4* __restrict__ p, long long n4) {
    long long i = (long long)blockIdx.x * blockDim.x + threadIdx.x;
    long long stride = (long long)gridDim.x * blockDim.x;
    float4 z = {0.f, 0.f, 0.f, 0.f};
    for (; i < n4; i += stride) p[i] = z;
}

// ---------------------------------------------------------------- degree counts
__global__ void k_deg(const int* __restrict__ sA, const int* __restrict__ dA,
                      const int* __restrict__ sB, const int* __restrict__ dB,
                      float* __restrict__ degAs, float* __restrict__ degAd,
                      float* __restrict__ degBs, float* __restrict__ degBd, int nE) {
    int e = blockIdx.x * blockDim.x + threadIdx.x;
    if (e >= nE) return;
    atomAddF(&degAs[sA[e]], 1.f);
    atomAddF(&degAd[dA[e]], 1.f);
    atomAddF(&degBs[sB[e]], 1.f);
    atomAddF(&degBd[dB[e]], 1.f);
}

// deg -> deg^-1/2 (0 if deg==0), in place over all 4 degree arrays (contiguous)
__global__ void k_inv(float* __restrict__ d, int n) {
    int i = blockIdx.x * blockDim.x + threadIdx.x;
    if (i >= n) return;
    float v = d[i];
    d[i] = (v > 0.f) ? rsqrtf(v) : 0.f;
}

// ---------------------------------------------------------------- edge scatter
// one wave (32 lanes) per edge; each lane handles 4 contiguous floats
__global__ void k_scatter(const float* __restrict__ xsrc,
                          const int* __restrict__ row, const int* __restrict__ col,
                          const float* __restrict__ inv_s, const float* __restrict__ inv_d,
                          float* __restrict__ s, float* __restrict__ sq, int nE) {
    int lane = threadIdx.x & 31;
    int wv   = threadIdx.x >> 5;
    int e = blockIdx.x * 8 + wv;
    if (e >= nE) return;
    int r = row[e], c = col[e];
    float w = inv_s[r] * inv_d[c];
    const float4 x = *(const float4*)(xsrc + (long long)r * DD + lane * 4);
    float4 m = {x.x * w, x.y * w, x.z * w, x.w * w};
    float* sp  = s  + (long long)c * DD + lane * 4;
    float* qp  = sq + (long long)c * DD + lane * 4;
    atomAddF(sp + 0, m.x); atomAddF(sp + 1, m.y); atomAddF(sp + 2, m.z); atomAddF(sp + 3, m.w);
    atomAddF(qp + 0, m.x * m.x); atomAddF(qp + 1, m.y * m.y);
    atomAddF(qp + 2, m.z * m.z); atomAddF(qp + 3, m.w * m.w);
}

// ---------------------------------------------------------------- fused triple-GEMM + bias + LayerNorm + ReLU
// block: 256 threads (8 waves). Block handles 16 rows; wave w handles columns [16w, 16w+16).
// out[n, :] = LN( X[n]@Wm^T + bm + S[n]@Wl^T + bl + FM[n]@Wf^T + bf ) -> ReLU
// FM = 0.5*(S*S - SQ)
__global__ void __launch_bounds__(256)
k_gemm_ln_relu(const float* __restrict__ X, const float* __restrict__ S,
               const float* __restrict__ SQ,
               const float* __restrict__ Wm, const float* __restrict__ bm,
               const float* __restrict__ Wl, const float* __restrict__ bl,
               const float* __restrict__ Wf, const float* __restrict__ bf,
               const float* __restrict__ g,  const float* __restrict__ be,
               float* __restrict__ out) {
    __shared__ float sh_h[16][LDP];
    __shared__ float sh_s[16][LDP];
    __shared__ float sh_f[16][LDP];
    __shared__ float sh_o[16][LDP];

    const int tid = threadIdx.x;
    const long long r0 = (long long)blockIdx.x * 16;

    // stage A tiles (coalesced): h, s, fm = 0.5*(s*s - sq)
    for (int i = tid; i < 16 * DD; i += 256) {
        int r = i >> 7, c = i & 127;
        long long gi = (r0 + r) * DD + c;
        float hv = X[gi];
        float sv = S[gi];
        float qv = SQ[gi];
        sh_h[r][c] = hv;
        sh_s[r][c] = sv;
        sh_f[r][c] = 0.5f * (sv * sv - qv);
    }
    __syncthreads();

    const int lane = tid & 31;
    const int wv   = tid >> 5;
    const int half = lane >> 4;   // 0: K pair {0,1}; 1: K pair {2,3} of each k-step
    const int l16  = lane & 15;   // A: row M; B: column N
    const int ncol = wv * 16 + l16;

    const float* wm = Wm + ncol * DD;
    const float* wl = Wl + ncol * DD;
    const float* wf = Wf + ncol * DD;

    v8f acc = {0.f, 0.f, 0.f, 0.f, 0.f, 0.f, 0.f, 0.f};

    // D = sum of three GEMMs: chain all WMMAs through one accumulator
    for (int k = 0; k < DD; k += 4) {
        int ka = k + 2 * half;
        v2f a, b;
        // X @ Wm^T
        a = *(const v2f*)(&sh_h[l16][ka]);
        b = *(const v2f*)(wm + ka);          // B[k][n] = W[n][k], contiguous in k
        acc = __builtin_amdgcn_wmma_f32_16x16x4_f32(false, a, false, b, (short)0, acc, false, false);
        // S @ Wl^T
        a = *(const v2f*)(&sh_s[l16][ka]);
        b = *(const v2f*)(wl + ka);
        acc = __builtin_amdgcn_wmma_f32_16x16x4_f32(false, a, false, b, (short)0, acc, false, false);
        // FM @ Wf^T
        a = *(const v2f*)(&sh_f[l16][ka]);
        b = *(const v2f*)(wf + ka);
        acc = __builtin_amdgcn_wmma_f32_16x16x4_f32(false, a, false, b, (short)0, acc, false, false);
    }

    // bias (per output column) and write accumulator to LDS
    // C/D layout: VGPR r -> (M=r, N=l16) for lanes 0-15, (M=r+8, N=l16) for lanes 16-31
    float bias = bm[ncol] + bl[ncol] + bf[ncol];
    #pragma unroll
    for (int r = 0; r < 8; ++r) {
        sh_o[r + 8 * half][ncol] = acc[r] + bias;
    }
    __syncthreads();

    // LayerNorm + ReLU: 16 threads per row, each owns 8 strided columns
    {
        int row  = tid >> 4;
        int part = tid & 15;
        float v[8];
        float sum = 0.f, ssq = 0.f;
        #pragma unroll
        for (int j = 0; j < 8; ++j) {
            float x = sh_o[row][part + 16 * j];
            v[j] = x;
            sum += x;
            ssq += x * x;
        }
        #pragma unroll
        for (int m = 1; m < 16; m <<= 1) {
            sum += __shfl_xor(sum, m, 16);
            ssq += __shfl_xor(ssq, m, 16);
        }
        float mu  = sum * (1.f / DD);
        float var = ssq * (1.f / DD) - mu * mu;
        float rin = rsqrtf(var + EPS);
        long long ob = (r0 + row) * DD;
        #pragma unroll
        for (int j = 0; j < 8; ++j) {
            int c = part + 16 * j;
            float y = (v[j] - mu) * rin * g[c] + be[c];
            out[ob + c] = fmaxf(y, 0.f);
        }
    }
}

extern "C" void kernel_launch(void* const* d_in, const int* in_sizes, int n_in,
                              void* d_out, int out_size, void* d_ws, size_t ws_size,
                              hipStream_t stream) {
    const float* h_user = (const float*)d_in[0];
    const float* h_item = (const float*)d_in[1];
    const int* u2i_src  = (const int*)d_in[2];
    const int* u2i_dst  = (const int*)d_in[3];
    const int* i2u_src  = (const int*)d_in[4];
    const int* i2u_dst  = (const int*)d_in[5];
    const float* Ws_user = (const float*)d_in[6];
    const float* bs_user = (const float*)d_in[7];
    const float* Ws_item = (const float*)d_in[8];
    const float* bs_item = (const float*)d_in[9];
    const float* Wlin_u2i = (const float*)d_in[10];
    const float* blin_u2i = (const float*)d_in[11];
    const float* Wfm_u2i  = (const float*)d_in[12];
    const float* bfm_u2i  = (const float*)d_in[13];
    const float* Wlin_i2u = (const float*)d_in[14];
    const float* blin_i2u = (const float*)d_in[15];
    const float* Wfm_i2u  = (const float*)d_in[16];
    const float* bfm_i2u  = (const float*)d_in[17];
    const float* g_user  = (const float*)d_in[18];
    const float* be_user = (const float*)d_in[19];
    const float* g_item  = (const float*)d_in[20];
    const float* be_item = (const float*)d_in[21];

    float* out_u = (float*)d_out;
    float* out_i = out_u + (long long)NU * DD;

    // workspace layout (floats)
    float* w = (float*)d_ws;
    float* degAs = w;                       // NU   (u2i src: users)
    float* degAd = degAs + NU;              // NI   (u2i dst: items)
    float* degBs = degAd + NI;              // NI   (i2u src: items)
    float* degBd = degBs + NI;              // NU   (i2u dst: users)
    float* s_i   = degBd + NU;              // NI*D
    float* sq_i  = s_i  + (long long)NI * DD;
    float* s_u   = sq_i + (long long)NI * DD;
    float* sq_u  = s_u  + (long long)NU * DD;

    const long long totalF = 2LL * (NU + NI) + 2LL * DD * (NU + NI);

    // 1) zero accumulators (atomics need clean state every call)
    k_zero<<<8192, 256, 0, stream>>>((float4*)w, totalF / 4);

    // 2) degree counts + inverse-sqrt transform
    k_deg<<<(NE + 255) / 256, 256, 0, stream>>>(u2i_src, u2i_dst, i2u_src, i2u_dst,
                                                degAs, degAd, degBs, degBd, NE);
    k_inv<<<(2 * (NU + NI) + 255) / 256, 256, 0, stream>>>(w, 2 * (NU + NI));

    // 3) edge scatter: messages -> s, sq (wave per edge, 8 edges per block)
    k_scatter<<<(NE + 7) / 8, 256, 0, stream>>>(h_user, u2i_src, u2i_dst,
                                                degAs, degAd, s_i, sq_i, NE);
    k_scatter<<<(NE + 7) / 8, 256, 0, stream>>>(h_item, i2u_src, i2u_dst,
                                                degBs, degBd, s_u, sq_u, NE);

    // 4) fused triple-GEMM (WMMA f32) + bias + LayerNorm + ReLU
    k_gemm_ln_relu<<<NI / 16, 256, 0, stream>>>(h_item, s_i, sq_i,
                                                Ws_item, bs_item,
                                                Wlin_u2i, blin_u2i,
                                                Wfm_u2i, bfm_u2i,
                                                g_item, be_item, out_i);
    k_gemm_ln_relu<<<NU / 16, 256, 0, stream>>>(h_user, s_u, sq_u,
                                                Ws_user, bs_user,
                                                Wlin_i2u, blin_i2u,
                                                Wfm_i2u, bfm_i2u,
                                                g_user, be_user, out_u);
}